// PyramidToDepth_82394652607163
// MI455X (gfx1250) — compile-verified
//
#include <hip/hip_runtime.h>

#define NLEV    9
#define HW      512
#define BATCH   8
#define ROWS    8
#define THREADS 256

#if __has_builtin(__builtin_amdgcn_global_load_async_to_lds_b128) && \
    __has_builtin(__builtin_amdgcn_s_wait_asynccnt)
#define USE_ASYNC_LDS 1
#else
#define USE_ASYNC_LDS 0
#endif

// Builtin parameter types (confirmed by hipcc diagnostic): pointers to int4
// vectors, AS1 (__device__) source, AS3 (__shared__) destination.
typedef int v4i_t __attribute__((ext_vector_type(4)));
typedef __attribute__((address_space(1))) v4i_t gv4i;
typedef __attribute__((address_space(3))) v4i_t lv4i;

// float2-element offset of pyramid level l (l>=1) inside the workspace.
__host__ __device__ constexpr size_t lvl_off(int l) {
    size_t o = 0;
    for (int k = 1; k < l; ++k) { int m = HW >> k; o += (size_t)BATCH * m * m; }
    return o;
}

// ---------------------------------------------------------------------------
// 2x2 average pool, compile-time output side M: src (BATCH,2M,2M) float2,
// dst (BATCH,M,M).  M is a power of two -> all index math folds to shifts.
// ---------------------------------------------------------------------------
template <int M>
__global__ __launch_bounds__(256)
void pool2x2_kernel(const float2* __restrict__ src, float2* __restrict__ dst) {
    constexpr int TOTAL = BATCH * M * M;
    int t = blockIdx.x * 256 + threadIdx.x;
    if (t >= TOTAL) return;
    int b = t / (M * M);
    int r = t % (M * M);
    int y = r / M;
    int x = r % M;
    constexpr int N = M * 2;
    const float2* s = src + (((size_t)b * N + 2 * y) * N + 2 * x);
    float2 a0 = s[0], a1 = s[1], a2 = s[N], a3 = s[N + 1];
    float2 o;
    o.x = (a0.x + a1.x + a2.x + a3.x) * 0.25f;
    o.y = (a0.y + a1.y + a2.y + a3.y) * 0.25f;
    dst[t] = o;
}

// ---------------------------------------------------------------------------
// Level 0: pure copy + layout transform, float4 (b128) loads and stores.
// ---------------------------------------------------------------------------
__device__ __forceinline__ void copy_level0(const float2* __restrict__ xin,
                                            float* __restrict__ out,
                                            unsigned b, unsigned rb, int tid) {
    const float2* src = xin + (size_t)b * HW * HW;
    for (int q = tid; q < ROWS * HW / 4; q += THREADS) {
        int y  = (int)rb * ROWS + (q >> 7);      // 128 quads per 512-px row
        int xq = (q & 127) << 2;
        const float2* p = src + (size_t)y * HW + xq;
        float2 a = p[0], c = p[1], d = p[2], e = p[3];
        size_t ob = (((size_t)b * (2 * NLEV)) * HW + y) * HW + xq;
        *(float4*)(out + ob)                     = make_float4(a.x, c.x, d.x, e.x);
        *(float4*)(out + ob + (size_t)HW * HW)   = make_float4(a.y, c.y, d.y, e.y);
    }
}

// ---------------------------------------------------------------------------
// Level LVL >= 1: bilinear upsample (half-pixel centers, edge clamp == JAX's
// renormalized linear kernel).  Levels with N <= 64 are staged whole into LDS
// via async global->LDS (ASYNCcnt) loads; coarse gathers then hit LDS.
// ---------------------------------------------------------------------------
template <int LVL>
__device__ __forceinline__ void upsample_level(const float2* __restrict__ ws,
                                               float* __restrict__ out,
                                               unsigned b, unsigned rb, int tid,
                                               float2* tile) {
    constexpr int  N       = HW >> LVL;
    constexpr bool LDS_FIT = (N <= 64);
    const float2* src = ws + lvl_off(LVL) + (size_t)b * N * N;

    if constexpr (LDS_FIT) {
        constexpr int CHUNKS = (N * N * (int)sizeof(float2)) / 16;
#if USE_ASYNC_LDS
        for (int c = tid; c < CHUNKS; c += THREADS) {
            __builtin_amdgcn_global_load_async_to_lds_b128(
                (gv4i*)((const char*)src + (size_t)c * 16),
                (lv4i*)((char*)tile + (size_t)c * 16),
                0, 0);
        }
        __builtin_amdgcn_s_wait_asynccnt(0);
#else
        for (int c = tid; c < CHUNKS; c += THREADS)
            ((float4*)tile)[c] = ((const float4*)src)[c];
#endif
        __syncthreads();
    }

    constexpr float scale = (float)N / (float)HW;

    for (int q = tid; q < ROWS * HW / 4; q += THREADS) {
        int y  = (int)rb * ROWS + (q >> 7);
        int xq = (q & 127) << 2;

        // row weights once per quad
        float sy = __builtin_fmaf((float)y + 0.5f, scale, -0.5f);
        float fy = floorf(sy);
        float wy = sy - fy;
        int  iy = (int)fy;
        int  y0 = iy < 0 ? 0 : iy;
        int  y1 = (iy + 1 > N - 1) ? (N - 1) : (iy + 1);

        const float2* row0;
        const float2* row1;
        if constexpr (LDS_FIT) {
            row0 = tile + y0 * N;
            row1 = tile + y1 * N;
        } else {
            row0 = src + (size_t)y0 * N;
            row1 = src + (size_t)y1 * N;
        }

        float r0[4], r1[4];
#pragma unroll
        for (int j = 0; j < 4; ++j) {
            int   xc = xq + j;
            float sx = __builtin_fmaf((float)xc + 0.5f, scale, -0.5f);
            float fx = floorf(sx);
            float wx = sx - fx;
            int   ix = (int)fx;
            int   x0 = ix < 0 ? 0 : ix;
            int   x1 = (ix + 1 > N - 1) ? (N - 1) : (ix + 1);
            float2 v00 = row0[x0], v01 = row0[x1];
            float2 v10 = row1[x0], v11 = row1[x1];
            float tx = __builtin_fmaf(wx, v01.x - v00.x, v00.x);
            float bx = __builtin_fmaf(wx, v11.x - v10.x, v10.x);
            r0[j] = __builtin_fmaf(wy, bx - tx, tx);
            float ty = __builtin_fmaf(wx, v01.y - v00.y, v00.y);
            float by = __builtin_fmaf(wx, v11.y - v10.y, v10.y);
            r1[j] = __builtin_fmaf(wy, by - ty, ty);
        }

        size_t ob = (((size_t)b * (2 * NLEV) + 2 * LVL) * HW + y) * HW + xq;
        *(float4*)(out + ob)                   = make_float4(r0[0], r0[1], r0[2], r0[3]);
        *(float4*)(out + ob + (size_t)HW * HW) = make_float4(r1[0], r1[1], r1[2], r1[3]);
    }
}

// ---------------------------------------------------------------------------
// One fused pass over the 151 MB output.  Block = (batch, level, 8-row stripe).
// ---------------------------------------------------------------------------
__global__ __launch_bounds__(THREADS)
void upsample_kernel(const float2* __restrict__ xin,
                     const float2* __restrict__ ws,
                     float* __restrict__ out) {
    __shared__ __align__(16) float2 tile[64 * 64];   // 32 KB, fits 320 KB/WGP easily

    unsigned blk = blockIdx.x;
    unsigned rb  = blk & 63u;
    unsigned tmp = blk >> 6;
    unsigned lvl = tmp % (unsigned)NLEV;
    unsigned b   = tmp / (unsigned)NLEV;
    int tid = threadIdx.x;

    switch (lvl) {
        case 0: copy_level0(xin, out, b, rb, tid); break;
        case 1: upsample_level<1>(ws, out, b, rb, tid, tile); break;
        case 2: upsample_level<2>(ws, out, b, rb, tid, tile); break;
        case 3: upsample_level<3>(ws, out, b, rb, tid, tile); break;
        case 4: upsample_level<4>(ws, out, b, rb, tid, tile); break;
        case 5: upsample_level<5>(ws, out, b, rb, tid, tile); break;
        case 6: upsample_level<6>(ws, out, b, rb, tid, tile); break;
        case 7: upsample_level<7>(ws, out, b, rb, tid, tile); break;
        default: upsample_level<8>(ws, out, b, rb, tid, tile); break;
    }
}

// ---------------------------------------------------------------------------
// Host launcher
// ---------------------------------------------------------------------------
extern "C" void kernel_launch(void* const* d_in, const int* in_sizes, int n_in,
                              void* d_out, int out_size, void* d_ws, size_t ws_size,
                              hipStream_t stream) {
    (void)in_sizes; (void)n_in; (void)out_size; (void)ws_size;

    const float2* x   = (const float2*)d_in[0];
    float2*       wsp = (float2*)d_ws;
    float*        out = (float*)d_out;

#define POOL_BLOCKS(M) ((BATCH * (M) * (M) + 255) / 256)
    pool2x2_kernel<256><<<POOL_BLOCKS(256), 256, 0, stream>>>(x,                 wsp + lvl_off(1));
    pool2x2_kernel<128><<<POOL_BLOCKS(128), 256, 0, stream>>>(wsp + lvl_off(1),  wsp + lvl_off(2));
    pool2x2_kernel< 64><<<POOL_BLOCKS( 64), 256, 0, stream>>>(wsp + lvl_off(2),  wsp + lvl_off(3));
    pool2x2_kernel< 32><<<POOL_BLOCKS( 32), 256, 0, stream>>>(wsp + lvl_off(3),  wsp + lvl_off(4));
    pool2x2_kernel< 16><<<POOL_BLOCKS( 16), 256, 0, stream>>>(wsp + lvl_off(4),  wsp + lvl_off(5));
    pool2x2_kernel<  8><<<POOL_BLOCKS(  8), 256, 0, stream>>>(wsp + lvl_off(5),  wsp + lvl_off(6));
    pool2x2_kernel<  4><<<POOL_BLOCKS(  4), 256, 0, stream>>>(wsp + lvl_off(6),  wsp + lvl_off(7));
    pool2x2_kernel<  2><<<POOL_BLOCKS(  2), 256, 0, stream>>>(wsp + lvl_off(7),  wsp + lvl_off(8));
#undef POOL_BLOCKS

    int grid = BATCH * NLEV * (HW / ROWS);   // 4608 blocks
    upsample_kernel<<<grid, THREADS, 0, stream>>>(x, wsp, out);
}